// ROIPooler_48404281426053
// MI455X (gfx1250) — compile-verified
//
#include <hip/hip_runtime.h>
#include <math.h>
#include <stdint.h>

typedef float v2f __attribute__((ext_vector_type(2)));
typedef float v8f __attribute__((ext_vector_type(8)));

// explicit global-address-space pointers so reconstructed bases lower to
// global_load/global_store (not flat_*, which also burns DScnt)
typedef const __attribute__((address_space(1))) float* gcfp;
typedef __attribute__((address_space(1))) float* gfp;

#define OUTW 7
#define NCH 256
#define NBOX 256
#define NBATCH 2
#define KMAX 128

__device__ __forceinline__ int imin(int a, int b) { return a < b ? a : b; }
__device__ __forceinline__ int imax(int a, int b) { return a > b ? a : b; }

struct RoiParams {
    uint32_t fblo, fbhi; // feature base (level + batch offset), split for readfirstlane
    int H;               // feature H == W (power of two)
    int wsh;             // log2(W)
    int xlo, ylo, Kx, Ky;
};

__global__ __launch_bounds__(256) void roi_pool_wmma(
    const float* __restrict__ x2, const float* __restrict__ x3,
    const float* __restrict__ x4, const float* __restrict__ x5,
    const float* __restrict__ boxes, float* __restrict__ out)
{
    __shared__ float WyL[KMAX][16];   // [y - ylo][oy]  (oy >= 7 stays zero)
    __shared__ float WxL[KMAX][16];   // [x - xlo][ox]
    __shared__ float Tl[8][16][17];   // per-wave D->A transpose scratch
    __shared__ RoiParams P;

    const int roi = blockIdx.x;
    const int tid = threadIdx.x;

    // zero the weight matrices
    for (int i = tid; i < KMAX * 16; i += 256) {
        ((float*)WyL)[i] = 0.0f;
        ((float*)WxL)[i] = 0.0f;
    }
    __syncthreads();

    if (tid == 0) {
        const float* bx = boxes + (size_t)roi * 4;
        float bx1 = bx[0], by1 = bx[1], bx2 = bx[2], by2 = bx[3];
        float area = (bx2 - bx1) * (by2 - by1);
        float size = sqrtf(fmaxf(area, 0.0f));
        float lf = floorf(4.0f + log2f(size / 224.0f + 1e-8f));
        lf = fminf(fmaxf(lf, 2.0f), 5.0f);
        int lvl = (int)lf - 2;                       // 0..3
        const float* feats[4] = { x2, x3, x4, x5 };
        int H = 256 >> lvl;
        float scale = 0.25f / (float)(1 << lvl);
        int b = roi / NBOX;
        const float* fb = feats[lvl] + ((size_t)b * NCH << (2 * (8 - lvl)));

        float sx1 = bx1 * scale - 0.5f, sx2 = bx2 * scale - 0.5f;
        float sy1 = by1 * scale - 0.5f, sy2 = by2 * scale - 0.5f;
        float bw = (sx2 - sx1) / (float)OUTW;
        float bh = (sy2 - sy1) / (float)OUTW;
        float Hf = (float)H;

        // ---- X weights ----
        int xlo = H - 1, xhi = 0;
        for (int j = 0; j < 14; ++j) {
            float t = sx1 + ((float)j + 0.5f) * 0.5f * bw;
            float tc = fminf(fmaxf(t, 0.0f), Hf - 1.0f);
            int i0 = (int)floorf(tc);
            xlo = imin(xlo, i0);
            xhi = imax(xhi, imin(i0 + 1, H - 1));
        }
        int Kx = imin(xhi - xlo + 1, KMAX);
        for (int j = 0; j < 14; ++j) {
            float t = sx1 + ((float)j + 0.5f) * 0.5f * bw;
            bool v = (t >= -1.0f) && (t <= Hf);
            float tc = fminf(fmaxf(t, 0.0f), Hf - 1.0f);
            int i0 = (int)floorf(tc);
            int i1 = imin(i0 + 1, H - 1);
            float l = tc - (float)i0;
            if (v) {
                int a = i0 - xlo, bcol = i1 - xlo;
                if (a >= 0 && a < KMAX)       WxL[a][j >> 1]    += 0.5f * (1.0f - l);
                if (bcol >= 0 && bcol < KMAX) WxL[bcol][j >> 1] += 0.5f * l;
            }
        }

        // ---- Y weights ----
        int ylo = H - 1, yhi = 0;
        for (int j = 0; j < 14; ++j) {
            float t = sy1 + ((float)j + 0.5f) * 0.5f * bh;
            float tc = fminf(fmaxf(t, 0.0f), Hf - 1.0f);
            int i0 = (int)floorf(tc);
            ylo = imin(ylo, i0);
            yhi = imax(yhi, imin(i0 + 1, H - 1));
        }
        int Ky = imin(yhi - ylo + 1, KMAX);
        for (int j = 0; j < 14; ++j) {
            float t = sy1 + ((float)j + 0.5f) * 0.5f * bh;
            bool v = (t >= -1.0f) && (t <= Hf);
            float tc = fminf(fmaxf(t, 0.0f), Hf - 1.0f);
            int i0 = (int)floorf(tc);
            int i1 = imin(i0 + 1, H - 1);
            float l = tc - (float)i0;
            if (v) {
                int a = i0 - ylo, brow = i1 - ylo;
                if (a >= 0 && a < KMAX)       WyL[a][j >> 1]    += 0.5f * (1.0f - l);
                if (brow >= 0 && brow < KMAX) WyL[brow][j >> 1] += 0.5f * l;
            }
        }

        uint64_t fb64 = (uint64_t)(uintptr_t)fb;
        P.fblo = (uint32_t)fb64;
        P.fbhi = (uint32_t)(fb64 >> 32);
        P.H = H; P.wsh = 8 - lvl;
        P.xlo = xlo; P.ylo = ylo; P.Kx = Kx; P.Ky = Ky;
    }
    __syncthreads();

    // ---- broadcast ROI params to SGPRs (uniform) ----
    const int H   = __builtin_amdgcn_readfirstlane(P.H);
    const int wsh = __builtin_amdgcn_readfirstlane(P.wsh);
    const int xlo = __builtin_amdgcn_readfirstlane(P.xlo);
    const int ylo = __builtin_amdgcn_readfirstlane(P.ylo);
    const int Kx  = __builtin_amdgcn_readfirstlane(P.Kx);
    const int Ky  = __builtin_amdgcn_readfirstlane(P.Ky);
    const uint32_t flo = __builtin_amdgcn_readfirstlane(P.fblo);
    const uint32_t fhi = __builtin_amdgcn_readfirstlane(P.fbhi);
    gcfp fb = (gcfp)(uintptr_t)(((uint64_t)fhi << 32) | flo);
    gfp  og = (gfp)(uintptr_t)out;

    const int Hm1 = H - 1;
    const int nXb = (Kx + 15) >> 4;
    const int nYc = (Ky + 3) >> 2;

    const int wv   = __builtin_amdgcn_readfirstlane(tid >> 5);  // wave id (scalar)
    const int lane = tid & 31;
    const int half = lane >> 4;     // 0 | 1
    const int l15  = lane & 15;
    const int k0   = half * 2;      // VGPR0 K offset: 0 | 2  (VGPR1: 1 | 3)

    for (int c = wv; c < NCH; c += 8) {
        gcfp fc = fb + ((size_t)c << (2 * wsh));                   // scalar base
        gfp  oc = og + ((size_t)roi * NCH + c) * (OUTW * OUTW);    // scalar base
        v8f acc = {};
        for (int nb = 0; nb < nXb; ++nb) {
            // ---- stage 1: T[oy, x] = sum_y Wy[oy,y] * f[y,x]  (coalesced row loads)
            v8f T = {};
            int colc = imin(xlo + (nb << 4) + l15, Hm1);
            #pragma unroll 2
            for (int t = 0; t < nYc; ++t) {
                int yb = ylo + (t << 2) + k0;
                int r0 = imin(yb,     Hm1);
                int r1 = imin(yb + 1, Hm1);
                v2f a, bm;
                a.x  = WyL[(t << 2) + k0][l15];       // A[m=l15, K=k0]
                a.y  = WyL[(t << 2) + k0 + 1][l15];   // A[m=l15, K=k0+1]
                bm.x = fc[(r0 << wsh) + colc];        // B[K=k0,   n=l15]
                bm.y = fc[(r1 << wsh) + colc];        // B[K=k0+1, n=l15]
                T = __builtin_amdgcn_wmma_f32_16x16x4_f32(
                        false, a, false, bm, (short)0, T, false, false);
            }
            // ---- D-layout -> A-layout via per-wave LDS tile
            #pragma unroll
            for (int r = 0; r < 8; ++r)
                Tl[wv][r + 8 * half][l15] = T[r];
            // (same-wave LDS is in-order; compiler inserts s_wait_dscnt before reuse)

            // ---- stage 2: acc[oy, ox] += sum_x T[oy,x] * Wx[x,ox]
            int nU = imin(4, (Kx - (nb << 4) + 3) >> 2);
            for (int u = 0; u < nU; ++u) {
                v2f a2, b2;
                a2.x = Tl[wv][l15][(u << 2) + k0];
                a2.y = Tl[wv][l15][(u << 2) + k0 + 1];
                int xr = (nb << 4) + (u << 2) + k0;
                b2.x = WxL[xr][l15];
                b2.y = WxL[xr + 1][l15];
                acc = __builtin_amdgcn_wmma_f32_16x16x4_f32(
                        false, a2, false, b2, (short)0, acc, false, false);
            }
        }
        // ---- write out: D layout -> out[roi, c, oy, ox]
        #pragma unroll
        for (int r = 0; r < 8; ++r) {
            int oy = r + 8 * half;
            int ox = l15;
            if (oy < OUTW && ox < OUTW)
                oc[oy * OUTW + ox] = acc[r];
        }
    }
}

extern "C" void kernel_launch(void* const* d_in, const int* in_sizes, int n_in,
                              void* d_out, int out_size, void* d_ws, size_t ws_size,
                              hipStream_t stream) {
    const float* x2 = (const float*)d_in[0];
    const float* x3 = (const float*)d_in[1];
    const float* x4 = (const float*)d_in[2];
    const float* x5 = (const float*)d_in[3];
    const float* boxes = (const float*)d_in[4];
    float* out = (float*)d_out;
    (void)in_sizes; (void)n_in; (void)out_size; (void)d_ws; (void)ws_size;
    roi_pool_wmma<<<NBATCH * NBOX, 256, 0, stream>>>(x2, x3, x4, x5, boxes, out);
}